// Conv3d_42279658062071
// MI455X (gfx1250) — compile-verified
//
#include <hip/hip_runtime.h>

typedef float v2f __attribute__((ext_vector_type(2)));
typedef float v4f __attribute__((ext_vector_type(4)));
typedef float v8f __attribute__((ext_vector_type(8)));

#define KVOL  27
#define MM    50000
#define CIN   64
#define COUT  64
#define MTILE 128      // rows of M per block: 8 waves x 16 rows
#define LDSP  66       // padded LDS pitch (words): stride-2 banks -> conflict-free

__global__ __launch_bounds__(256)
void zero_out_kernel(float* __restrict__ out, int n4) {
    int i = blockIdx.x * blockDim.x + threadIdx.x;
    if (i < n4) ((v4f*)out)[i] = (v4f){0.f, 0.f, 0.f, 0.f};
}

__global__ __launch_bounds__(256)
void sparse_conv_wmma(const float* __restrict__ feats,
                      const float* __restrict__ kern,
                      const int*   __restrict__ in_idx,
                      const int*   __restrict__ out_idx,
                      float*       __restrict__ out) {
    // kernel[k] slice staged transposed: ldsB[cout * LDSP + cin]
    __shared__ float ldsB[COUT * LDSP];

    const int k    = blockIdx.y;
    const int tid  = threadIdx.x;
    const int wave = tid >> 5;
    const int lane = tid & 31;
    const int half = lane >> 4;   // 0: lanes 0-15, 1: lanes 16-31
    const int l    = lane & 15;

    // ---- stage 64x64 f32 weight slice into LDS (transposed, padded) ----
    const v4f* ksrc = (const v4f*)(kern + (size_t)k * CIN * COUT);
    #pragma unroll
    for (int i = tid; i < (CIN * COUT) / 4; i += 256) {
        v4f v = ksrc[i];
        int f   = i * 4;
        int cin = f >> 6;    // row in weight slice
        int co  = f & 63;    // col
        ldsB[(co + 0) * LDSP + cin] = v.x;
        ldsB[(co + 1) * LDSP + cin] = v.y;
        ldsB[(co + 2) * LDSP + cin] = v.z;
        ldsB[(co + 3) * LDSP + cin] = v.w;
    }
    __syncthreads();

    // ---- per-lane gather base for the A fragment ----
    const int  m0      = blockIdx.x * MTILE + wave * 16;
    const int  mrow    = m0 + l;                 // A row this lane carries
    const bool valid_a = (mrow < MM);
    int aidx = 0;
    if (valid_a) aidx = in_idx[k * MM + mrow];
    const float* abase = feats + (size_t)aidx * CIN + 2 * half;

    // B fragment base pointers in LDS, one per 16-col tile
    const float* b0 = &ldsB[( 0 + l) * LDSP + 2 * half];
    const float* b1 = &ldsB[(16 + l) * LDSP + 2 * half];
    const float* b2 = &ldsB[(32 + l) * LDSP + 2 * half];
    const float* b3 = &ldsB[(48 + l) * LDSP + 2 * half];

    v8f acc[4] = {v8f{}, v8f{}, v8f{}, v8f{}};

    // ---- K loop: C_IN = 64 in 16 steps of K=4 ----
    #pragma unroll 4
    for (int kk = 0; kk < 16; ++kk) {
        v2f a = {0.f, 0.f};
        if (valid_a) a = *(const v2f*)(abase + 4 * kk);   // 8B gather per lane
        v2f f0 = *(const v2f*)(b0 + 4 * kk);              // ds_load_b64, no conflicts
        v2f f1 = *(const v2f*)(b1 + 4 * kk);
        v2f f2 = *(const v2f*)(b2 + 4 * kk);
        v2f f3 = *(const v2f*)(b3 + 4 * kk);
        acc[0] = __builtin_amdgcn_wmma_f32_16x16x4_f32(false, a, false, f0, (short)0, acc[0], false, false);
        acc[1] = __builtin_amdgcn_wmma_f32_16x16x4_f32(false, a, false, f1, (short)0, acc[1], false, false);
        acc[2] = __builtin_amdgcn_wmma_f32_16x16x4_f32(false, a, false, f2, (short)0, acc[2], false, false);
        acc[3] = __builtin_amdgcn_wmma_f32_16x16x4_f32(false, a, false, f3, (short)0, acc[3], false, false);
    }

    // ---- scatter-accumulate: VGPR r holds row (r + 8*half), col = 16n + l ----
    #pragma unroll
    for (int r = 0; r < 8; ++r) {
        int m = m0 + r + 8 * half;
        if (m < MM) {
            int oidx = out_idx[k * MM + m];
            float* orow = out + (size_t)oidx * COUT + l;
            atomicAdd(orow +  0, acc[0][r]);
            atomicAdd(orow + 16, acc[1][r]);
            atomicAdd(orow + 32, acc[2][r]);
            atomicAdd(orow + 48, acc[3][r]);
        }
    }
}

extern "C" void kernel_launch(void* const* d_in, const int* in_sizes, int n_in,
                              void* d_out, int out_size, void* d_ws, size_t ws_size,
                              hipStream_t stream) {
    const float* feats   = (const float*)d_in[0];
    const float* kern    = (const float*)d_in[1];
    const int*   in_idx  = (const int*)d_in[2];
    const int*   out_idx = (const int*)d_in[3];
    float*       out     = (float*)d_out;

    int n4 = out_size / 4;   // out_size = N * COUT, multiple of 4
    zero_out_kernel<<<(n4 + 255) / 256, 256, 0, stream>>>(out, n4);

    dim3 grid((MM + MTILE - 1) / MTILE, KVOL);
    sparse_conv_wmma<<<grid, 256, 0, stream>>>(feats, kern, in_idx, out_idx, out);
}